// Attention_76433238000364
// MI455X (gfx1250) — compile-verified
//
#include <hip/hip_runtime.h>
#include <hip/hip_bf16.h>

#define B_   128
#define L_   1024
#define RNN_ 1024
#define ATT_ 512
#define LSPLIT 4
#define LCHUNK (L_ / LSPLIT)   // 256

typedef __attribute__((ext_vector_type(2)))  float    v2f;
typedef __attribute__((ext_vector_type(4)))  float    v4f;
typedef __attribute__((ext_vector_type(8)))  float    v8f;

// ---------------------------------------------------------------------------
// Kernel 1: att_h[B,ATT] = h[B,RNN] @ W[ATT,RNN]^T + bias
// One wave (32 lanes) computes one 16x16 C tile via V_WMMA_F32_16X16X4_F32.
// A tile (16x4 of h):  lane r(0..15)/half hf: a = h[m0+r][k0+2hf .. +1]
// B tile (4x16 = W^T): lane r(0..15)/half hf: b = W[n0+r][k0+2hf .. +1]
// C layout: VGPR i, lane -> row m0 + i + 8*half, col n0 + (lane&15)
// ---------------------------------------------------------------------------
__global__ __launch_bounds__(32) void k_h2att(const float* __restrict__ h,
                                              const float* __restrict__ W,
                                              const float* __restrict__ bias,
                                              float* __restrict__ att_h)
{
    const int m0   = blockIdx.x * 16;
    const int n0   = blockIdx.y * 16;
    const int lane = threadIdx.x;
    const int half = lane >> 4;
    const int r    = lane & 15;

    const float* hrow = h + (size_t)(m0 + r) * RNN_;
    const float* wrow = W + (size_t)(n0 + r) * RNN_;

    v8f c = {};
    for (int k0 = 0; k0 < RNN_; k0 += 4) {
        v2f a = *(const v2f*)(hrow + k0 + 2 * half);
        v2f b = *(const v2f*)(wrow + k0 + 2 * half);
        c = __builtin_amdgcn_wmma_f32_16x16x4_f32(
                /*neg_a=*/false, a, /*neg_b=*/false, b,
                /*c_mod=*/(short)0, c, /*reuse_a=*/false, /*reuse_b=*/false);
    }

    const float bn = bias[n0 + r];
    #pragma unroll
    for (int i = 0; i < 8; ++i) {
        const int m = m0 + i + 8 * half;
        att_h[(size_t)m * ATT_ + n0 + r] = c[i] + bn;
    }
}

// ---------------------------------------------------------------------------
// Kernel 2: dot[b,l] = sum_a tanh(f2[b,l,a] + att_h[b,a]) * w_alpha[a] + b_alpha
// One wave per (b,l); 4x float4 NT loads per lane; wave32 xor-shuffle reduce.
// ---------------------------------------------------------------------------
__global__ __launch_bounds__(256) void k_score(const float* __restrict__ f2,
                                               const float* __restrict__ att_h,
                                               const float* __restrict__ w_alpha,
                                               const float* __restrict__ b_alpha,
                                               float* __restrict__ dot)
{
    const int b    = blockIdx.y;
    const int wave = threadIdx.x >> 5;
    const int lane = threadIdx.x & 31;
    const int l    = blockIdx.x * 8 + wave;

    const float* row = f2    + ((size_t)b * L_ + l) * ATT_;
    const float* ah  = att_h + (size_t)b * ATT_;

    float s = 0.0f;
    #pragma unroll
    for (int k = 0; k < 4; ++k) {
        const int a = (lane + 32 * k) * 4;           // coalesced 512B/load/wave
        v4f x  = __builtin_nontemporal_load((const v4f*)(row + a));
        v4f hh = *(const v4f*)(ah + a);
        v4f wa = *(const v4f*)(w_alpha + a);
        s += tanhf(x[0] + hh[0]) * wa[0];
        s += tanhf(x[1] + hh[1]) * wa[1];
        s += tanhf(x[2] + hh[2]) * wa[2];
        s += tanhf(x[3] + hh[3]) * wa[3];
    }
    #pragma unroll
    for (int off = 16; off > 0; off >>= 1)
        s += __shfl_xor(s, off, 32);

    if (lane == 0)
        dot[(size_t)b * L_ + l] = s + b_alpha[0];
}

// ---------------------------------------------------------------------------
// Kernel 3: weight[b,l] = exp(dot-max)*mask / sum_l(exp(dot-max)*mask)
// (softmax -> *mask -> renorm collapses to a single normalization pass)
// ---------------------------------------------------------------------------
__global__ __launch_bounds__(256) void k_softmax(const float* __restrict__ dot,
                                                 const float* __restrict__ mask,
                                                 float* __restrict__ weight)
{
    __shared__ float red[256];
    const int b = blockIdx.x;
    const int t = threadIdx.x;

    float v[4];
    float m = -3.402823466e+38f;
    #pragma unroll
    for (int i = 0; i < 4; ++i) {
        v[i] = dot[(size_t)b * L_ + t + 256 * i];
        m = fmaxf(m, v[i]);
    }
    red[t] = m; __syncthreads();
    for (int off = 128; off > 0; off >>= 1) {
        if (t < off) red[t] = fmaxf(red[t], red[t + off]);
        __syncthreads();
    }
    const float M = red[0];
    __syncthreads();

    float e[4];
    float s = 0.0f;
    #pragma unroll
    for (int i = 0; i < 4; ++i) {
        e[i] = __expf(v[i] - M) * mask[(size_t)b * L_ + t + 256 * i];
        s += e[i];
    }
    red[t] = s; __syncthreads();
    for (int off = 128; off > 0; off >>= 1) {
        if (t < off) red[t] += red[t + off];
        __syncthreads();
    }
    const float inv = 1.0f / red[0];

    #pragma unroll
    for (int i = 0; i < 4; ++i)
        weight[(size_t)b * L_ + t + 256 * i] = e[i] * inv;
}

// ---------------------------------------------------------------------------
// Kernel 4a: partial[ls][b][d] = sum_{l in chunk ls} weight[b,l] * f1[b,l,d]
// Block (b, ls): 256 threads x float4 cover the full 1024-wide row;
// global_load_b128 NT streaming (512 B per wave per load), weights in LDS.
// Deterministic partials (no float atomics) -> bit-stable across replays.
// ---------------------------------------------------------------------------
__global__ __launch_bounds__(256) void k_wsum_part(const float* __restrict__ f1,
                                                   const float* __restrict__ weight,
                                                   float* __restrict__ partial)
{
    __shared__ float wl[LCHUNK];
    const int b  = blockIdx.x;
    const int ls = blockIdx.y;            // 0..LSPLIT-1
    const int t  = threadIdx.x;
    const int l0 = ls * LCHUNK;

    wl[t] = weight[(size_t)b * L_ + l0 + t];
    __syncthreads();

    const float* base = f1 + ((size_t)b * L_ + l0) * RNN_ + t * 4;
    v4f acc = {0.0f, 0.0f, 0.0f, 0.0f};
    #pragma unroll 4
    for (int l = 0; l < LCHUNK; ++l) {
        // Speculative prefetch; past-the-end addresses are silently dropped.
        __builtin_prefetch(base + (size_t)(l + 8) * RNN_, 0, 0);
        v4f x = __builtin_nontemporal_load((const v4f*)(base + (size_t)l * RNN_));
        const float w = wl[l];
        acc[0] = fmaf(w, x[0], acc[0]);
        acc[1] = fmaf(w, x[1], acc[1]);
        acc[2] = fmaf(w, x[2], acc[2]);
        acc[3] = fmaf(w, x[3], acc[3]);
    }
    *(v4f*)(partial + ((size_t)ls * B_ + b) * RNN_ + t * 4) = acc;
}

// ---------------------------------------------------------------------------
// Kernel 4b: out[b,d] = sum_ls partial[ls][b][d]   (trivial 2 MB reduce)
// ---------------------------------------------------------------------------
__global__ __launch_bounds__(256) void k_reduce(const float* __restrict__ partial,
                                                float* __restrict__ out)
{
    const int b = blockIdx.x;
    const int t = threadIdx.x;
    const size_t off = (size_t)b * RNN_ + t * 4;

    v4f s = *(const v4f*)(partial + off);
    #pragma unroll
    for (int ls = 1; ls < LSPLIT; ++ls) {
        v4f p = *(const v4f*)(partial + (size_t)ls * B_ * RNN_ + off);
        s[0] += p[0]; s[1] += p[1]; s[2] += p[2]; s[3] += p[3];
    }
    *(v4f*)(out + off) = s;
}

// ---------------------------------------------------------------------------
extern "C" void kernel_launch(void* const* d_in, const int* in_sizes, int n_in,
                              void* d_out, int out_size, void* d_ws, size_t ws_size,
                              hipStream_t stream)
{
    (void)in_sizes; (void)n_in; (void)out_size; (void)ws_size;

    const float* h        = (const float*)d_in[0];   // [B,RNN]
    const float* f1       = (const float*)d_in[1];   // [B,L,RNN]
    const float* f2       = (const float*)d_in[2];   // [B,L,ATT]
    const float* masks    = (const float*)d_in[3];   // [B,L]
    const float* W_h2att  = (const float*)d_in[4];   // [ATT,RNN]
    const float* b_h2att  = (const float*)d_in[5];   // [ATT]
    const float* w_alpha  = (const float*)d_in[6];   // [ATT]
    const float* b_alpha  = (const float*)d_in[7];   // [1]
    float*       out      = (float*)d_out;           // [B,RNN]

    float* ws      = (float*)d_ws;
    float* att_h   = ws;                                  // B*ATT        =  65536 f
    float* dot     = att_h  + (size_t)B_ * ATT_;          // B*L          = 131072 f
    float* weight  = dot    + (size_t)B_ * L_;            // B*L          = 131072 f
    float* partial = weight + (size_t)B_ * L_;            // LSPLIT*B*RNN = 524288 f
                                                          // total ~3.25 MB of ws

    k_h2att    <<<dim3(B_ / 16, ATT_ / 16), 32,  0, stream>>>(h, W_h2att, b_h2att, att_h);
    k_score    <<<dim3(L_ / 8, B_),         256, 0, stream>>>(f2, att_h, w_alpha, b_alpha, dot);
    k_softmax  <<<B_,                       256, 0, stream>>>(dot, masks, weight);
    k_wsum_part<<<dim3(B_, LSPLIT),         256, 0, stream>>>(f1, weight, partial);
    k_reduce   <<<B_,                       256, 0, stream>>>(partial, out);
}